// GATLayer_81698867904518
// MI455X (gfx1250) — compile-verified
//
#include <hip/hip_runtime.h>
#include <hip/hip_bf16.h>
#include <stdint.h>

#define NEG_SLOPE 0.2f
#define EPS_GAT 1e-10f

typedef float v2f __attribute__((ext_vector_type(2)));
typedef float v8f __attribute__((ext_vector_type(8)));

// ---- monotone float <-> uint encoding so segment-max can use GLOBAL_ATOMIC_MAX_U32
__device__ __forceinline__ unsigned enc_ord(float f) {
  unsigned u = __float_as_uint(f);
  return (u & 0x80000000u) ? ~u : (u | 0x80000000u);
}
__device__ __forceinline__ float dec_ord(unsigned u) {
  return __uint_as_float((u & 0x80000000u) ? (u & 0x7FFFFFFFu) : ~u);
}

// ---- zero output, init segment-max to enc(-inf), segment-sum to 0
__global__ void gat_init(float* __restrict__ out, unsigned* __restrict__ maxb,
                         float* __restrict__ sumb, int outN, int segN) {
  int i = blockIdx.x * blockDim.x + threadIdx.x;
  if (i < outN) out[i] = 0.0f;
  if (i < segN) { maxb[i] = 0x007FFFFFu; /* enc(-inf) */ sumb[i] = 0.0f; }
}

// ---- h = x @ W^T with V_WMMA_F32_16X16X4_F32
// One wave owns TWO 16-row A tiles (32 rows) x all 8 column tiles:
//   - each B fragment feeds 2 WMMAs (halves W-fragment traffic)
//   - software-pipelined: next-k fragments prefetched into distinct regs while
//     current WMMAs issue -> partial s_wait_loadcnt instead of full waits.
// __launch_bounds__(256, 1): ~180 live VGPRs (128 acc + 40 frags + addressing);
// default occupancy budget (~128) caused accumulator spills to scratch. This
// kernel is bandwidth-trivial; occupancy is irrelevant, spills are not.
__global__ void __launch_bounds__(256, 1)
gat_gemm(const float* __restrict__ x, const float* __restrict__ W,
         float* __restrict__ hout, int M) {
  int wave = (int)((blockIdx.x * blockDim.x + threadIdx.x) >> 5);
  int lane = threadIdx.x & 31;
  int m0 = wave * 32;
  if (m0 >= M) return;
  int r = lane & 15, hi = lane >> 4;

  // A 16x4 layout: lane (r,hi) holds X[row, k+2hi], X[row, k+2hi+1]  (contiguous float2)
  // B = W^T so B[k, n0+r] = W[n0+r, k]: same contiguous-float2 addressing on W rows.
  const float* x0 = x + (size_t)(m0 + r) * 128 + 2 * hi;        // rows m0 .. m0+15
  const float* x1 = x0 + (size_t)16 * 128;                      // rows m0+16 .. m0+31
  const float* w0 = W + (size_t)r * 128 + 2 * hi;               // col tile t: + t*16*128

  v8f acc0[8] = {};
  v8f acc1[8] = {};

  v2f a0c, a1c, bc[8];
  v2f a0n, a1n, bn[8];

  // prologue: fragments for k = 0
  a0c = *(const v2f*)(x0);
  a1c = *(const v2f*)(x1);
#pragma unroll
  for (int t = 0; t < 8; ++t) bc[t] = *(const v2f*)(w0 + (size_t)t * 2048);

#pragma unroll
  for (int k = 0; k < 128; k += 4) {
    if (k + 4 < 128) {  // prefetch next k-step into distinct registers
      a0n = *(const v2f*)(x0 + k + 4);
      a1n = *(const v2f*)(x1 + k + 4);
#pragma unroll
      for (int t = 0; t < 8; ++t)
        bn[t] = *(const v2f*)(w0 + (size_t)t * 2048 + k + 4);
    }
#pragma unroll
    for (int t = 0; t < 8; ++t)
      acc0[t] = __builtin_amdgcn_wmma_f32_16x16x4_f32(false, a0c, false, bc[t],
                                                      (short)0, acc0[t], false, false);
#pragma unroll
    for (int t = 0; t < 8; ++t)
      acc1[t] = __builtin_amdgcn_wmma_f32_16x16x4_f32(false, a1c, false, bc[t],
                                                      (short)0, acc1[t], false, false);
    a0c = a0n;
    a1c = a1n;
#pragma unroll
    for (int t = 0; t < 8; ++t) bc[t] = bn[t];
  }

  // C/D layout: VGPR g, lane (r,hi) -> row base + g + 8*hi, col t*16 + r
#pragma unroll
  for (int t = 0; t < 8; ++t)
#pragma unroll
    for (int g = 0; g < 8; ++g) {
      hout[(size_t)(m0 + g + 8 * hi) * 128 + t * 16 + r]      = acc0[t][g];
      hout[(size_t)(m0 + 16 + g + 8 * hi) * 128 + t * 16 + r] = acc1[t][g];
    }
}

// ---- per-node logits: wave per (node,head), lane = d, wave32 butterfly reduce
__global__ void gat_proj(const float* __restrict__ hbuf, const float* __restrict__ a_src,
                         const float* __restrict__ a_dst, float* __restrict__ esrc,
                         float* __restrict__ edst, int BN) {
  int wave = (int)((blockIdx.x * blockDim.x + threadIdx.x) >> 5);
  int lane = threadIdx.x & 31;
  int bn = wave >> 2, h = wave & 3;
  if (bn >= BN) return;
  float hv = hbuf[(size_t)bn * 128 + h * 32 + lane];
  float s = hv * a_src[h * 32 + lane];
  float t = hv * a_dst[h * 32 + lane];
#pragma unroll
  for (int off = 16; off > 0; off >>= 1) {
    s += __shfl_xor(s, off, 32);
    t += __shfl_xor(t, off, 32);
  }
  if (lane == 0) { esrc[bn * 4 + h] = s; edst[bn * 4 + h] = t; }
}

__device__ __forceinline__ float edge_score(const float* esrc, const float* edst,
                                            size_t sidx, size_t didx, int h) {
  float sc = esrc[sidx * 4 + h] + edst[didx * 4 + h];
  return sc > 0.f ? sc : NEG_SLOPE * sc;
}

// ---- pass 1: segment max over dst via ordered-uint atomicMax
__global__ void gat_edge_max(const int* __restrict__ src, const int* __restrict__ dst,
                             const float* __restrict__ esrc, const float* __restrict__ edst,
                             unsigned* __restrict__ maxb, int E_, int B_, int N_) {
  int i = blockIdx.x * blockDim.x + threadIdx.x;
  if (i >= E_ * B_ * 4) return;
  int h = i & 3;
  int bi = (i >> 2) % B_;
  int e = (i >> 2) / B_;
  size_t sidx = (size_t)bi * N_ + src[e];
  size_t didx = (size_t)bi * N_ + dst[e];
  float sc = edge_score(esrc, edst, sidx, didx, h);
  atomicMax(&maxb[didx * 4 + h], enc_ord(sc));
}

// ---- pass 2: segment sum of exp(score - max)
__global__ void gat_edge_sum(const int* __restrict__ src, const int* __restrict__ dst,
                             const float* __restrict__ esrc, const float* __restrict__ edst,
                             const unsigned* __restrict__ maxb, float* __restrict__ sumb,
                             int E_, int B_, int N_) {
  int i = blockIdx.x * blockDim.x + threadIdx.x;
  if (i >= E_ * B_ * 4) return;
  int h = i & 3;
  int bi = (i >> 2) % B_;
  int e = (i >> 2) / B_;
  size_t sidx = (size_t)bi * N_ + src[e];
  size_t didx = (size_t)bi * N_ + dst[e];
  float sc = edge_score(esrc, edst, sidx, didx, h);
  atomicAdd(&sumb[didx * 4 + h], expf(sc - dec_ord(maxb[didx * 4 + h])));
}

// ---- pass 3: wave per (edge,batch,head); lane = d; coalesced 128B gather + atomic add
__global__ void gat_edge_agg(const int* __restrict__ src, const int* __restrict__ dst,
                             const float* __restrict__ esrc, const float* __restrict__ edst,
                             const unsigned* __restrict__ maxb, const float* __restrict__ sumb,
                             const float* __restrict__ hbuf, float* __restrict__ out,
                             int E_, int B_, int N_) {
  long long w = ((long long)blockIdx.x * blockDim.x + threadIdx.x) >> 5;
  int lane = threadIdx.x & 31;
  if (w >= (long long)E_ * B_ * 4) return;
  int h = (int)(w & 3);
  long long t = w >> 2;
  int bi = (int)(t % B_);
  long long e = t / B_;
  size_t sidx = (size_t)bi * N_ + src[e];
  size_t didx = (size_t)bi * N_ + dst[e];
  float sc = edge_score(esrc, edst, sidx, didx, h);
  float alpha = expf(sc - dec_ord(maxb[didx * 4 + h])) / (sumb[didx * 4 + h] + EPS_GAT);
  float v = alpha * hbuf[sidx * 128 + h * 32 + lane];
  atomicAdd(&out[didx * 128 + h * 32 + lane], v);
}

extern "C" void kernel_launch(void* const* d_in, const int* in_sizes, int n_in,
                              void* d_out, int out_size, void* d_ws, size_t ws_size,
                              hipStream_t stream) {
  (void)in_sizes; (void)n_in; (void)out_size; (void)ws_size;
  const float* x     = (const float*)d_in[0];
  const int*   ei    = (const int*)d_in[1];   // edge_index: [0,:]=src, [1,:]=dst
  const float* W     = (const float*)d_in[2];
  const float* a_src = (const float*)d_in[3];
  const float* a_dst = (const float*)d_in[4];
  float* out = (float*)d_out;

  const int B = 2, N = 20000, E = 640000;
  const int BN = B * N;

  // workspace carve-up (~23.04 MB total)
  float*    hbuf = (float*)d_ws;                       // BN*128
  float*    esrc = hbuf + (size_t)BN * 128;            // BN*4
  float*    edst = esrc + (size_t)BN * 4;              // BN*4
  unsigned* maxb = (unsigned*)(edst + (size_t)BN * 4); // BN*4
  float*    sumb = (float*)(maxb + (size_t)BN * 4);    // BN*4

  {
    int n = BN * 128;
    gat_init<<<(n + 255) / 256, 256, 0, stream>>>(out, maxb, sumb, n, BN * 4);
  }
  {
    int waves = BN / 32;  // 1250 waves, 32 rows each
    gat_gemm<<<(waves * 32 + 255) / 256, 256, 0, stream>>>(x, W, hbuf, BN);
  }
  {
    int waves = BN * 4;   // one wave per (node, head)
    gat_proj<<<(waves * 32 + 255) / 256, 256, 0, stream>>>(hbuf, a_src, a_dst, esrc, edst, BN);
  }
  {
    int n = E * B * 4;    // thread per (edge, batch, head)
    gat_edge_max<<<(n + 255) / 256, 256, 0, stream>>>(ei, ei + E, esrc, edst, maxb, E, B, N);
    gat_edge_sum<<<(n + 255) / 256, 256, 0, stream>>>(ei, ei + E, esrc, edst, maxb, sumb, E, B, N);
    long long threads = (long long)n * 32;  // wave per (edge, batch, head)
    gat_edge_agg<<<(unsigned)((threads + 255) / 256), 256, 0, stream>>>(
        ei, ei + E, esrc, edst, maxb, sumb, hbuf, out, E, B, N);
  }
}